// Net_5050881540298
// MI455X (gfx1250) — compile-verified
//
#include <hip/hip_runtime.h>
#include <hip/hip_bf16.h>

#define N_NODES   100000
#define N_EDGES   3200000
#define N_FEAT    512
#define HIDDEN    16
#define N_CLASSES 64
#define N_TILES   (N_NODES / 16)   // 6250 exactly: every valid tile is full

typedef __attribute__((ext_vector_type(2))) float v2f;
typedef __attribute__((ext_vector_type(8))) float v8f;

// ---------------------------------------------------------------------------
// 0) init: deg = 1.0 (self-loop weight), a1 = 0, a2 = 0
// ---------------------------------------------------------------------------
__global__ __launch_bounds__(256) void init_kernel(float* __restrict__ deg,
                                                   float* __restrict__ a1,
                                                   float* __restrict__ a2) {
    int i = blockIdx.x * 256 + threadIdx.x;           // up to N_NODES*64
    if (i < N_NODES) deg[i] = 1.0f;
    if (i < N_NODES * HIDDEN) a1[i] = 0.0f;
    if (i < N_NODES * N_CLASSES) a2[i] = 0.0f;
}

// ---------------------------------------------------------------------------
// 1) degree accumulation over target (col) nodes
// ---------------------------------------------------------------------------
__global__ __launch_bounds__(256) void deg_kernel(const int* __restrict__ ecol,
                                                  const float* __restrict__ ew,
                                                  float* __restrict__ deg) {
    int e = blockIdx.x * 256 + threadIdx.x;
    if (e < N_EDGES) atomicAdd(&deg[ecol[e]], ew[e]);
}

// ---------------------------------------------------------------------------
// 2) dinv = deg > 0 ? rsqrt(deg) : 0
// ---------------------------------------------------------------------------
__global__ __launch_bounds__(256) void dinv_kernel(const float* __restrict__ deg,
                                                   float* __restrict__ dinv) {
    int i = blockIdx.x * 256 + threadIdx.x;
    if (i < N_NODES) {
        float d = deg[i];
        dinv[i] = (d > 0.0f) ? rsqrtf(d) : 0.0f;
    }
}

// ---------------------------------------------------------------------------
// 3) h1 = x @ W1   via V_WMMA_F32_16X16X4_F32, one 16-row tile per wave32.
//    W1 (512x16 = 32KB) staged in LDS per block of 8 waves.
//    N_NODES % 16 == 0 -> whole-wave guard only; tile body is unguarded.
// ---------------------------------------------------------------------------
__global__ __launch_bounds__(256) void gemm1_kernel(const float* __restrict__ x,
                                                    const float* __restrict__ W1,
                                                    float* __restrict__ h1) {
    __shared__ float sW[N_FEAT * HIDDEN];
    int tid = threadIdx.x;
    for (int i = tid; i < N_FEAT * HIDDEN; i += 256) sW[i] = W1[i];
    __syncthreads();

    int wave = tid >> 5;
    int lane = tid & 31;
    int tile = blockIdx.x * 8 + wave;                 // 16-row output tile
    if (tile >= N_TILES) return;                      // per-wave guard only
    int rowBase = tile * 16;
    int laneHalf = lane >> 4;                         // 0: lanes 0-15, 1: 16-31
    int laneM = lane & 15;                            // A: row in tile, B: col
    const float* xrow = x + (size_t)(rowBase + laneM) * N_FEAT;

    int kOff = laneHalf * 2;                          // lanes 16-31 hold K+2,K+3
    v8f c = {};
    for (int k = 0; k < N_FEAT; k += 4) {
        v2f a, b;
        a.x = xrow[k + kOff];                         // A[M=laneM][k+kOff]
        a.y = xrow[k + kOff + 1];
        b.x = sW[(k + kOff) * HIDDEN + laneM];        // B[k+kOff][N=laneM]
        b.y = sW[(k + kOff + 1) * HIDDEN + laneM];
        c = __builtin_amdgcn_wmma_f32_16x16x4_f32(false, a, false, b,
                                                  (short)0, c, false, false);
    }
    // C/D layout: VGPR r -> M = r + laneHalf*8, N = laneM
    float* orow = h1 + (size_t)(rowBase + laneHalf * 8) * HIDDEN + laneM;
    #pragma unroll
    for (int r = 0; r < 8; ++r) orow[r * HIDDEN] = c[r];
}

// ---------------------------------------------------------------------------
// 4) scatter layer-1 messages: a1[col] += norm * h1[row]; 16 lanes per edge
// ---------------------------------------------------------------------------
__global__ __launch_bounds__(256) void scatter1_kernel(const int* __restrict__ ei,
                                                       const float* __restrict__ ew,
                                                       const float* __restrict__ dinv,
                                                       const float* __restrict__ h1,
                                                       float* __restrict__ a1) {
    int t = blockIdx.x * 256 + threadIdx.x;
    int e = t >> 4;
    int j = t & 15;
    if (e >= N_EDGES) return;
    int r = ei[e];                 // source (row)
    int c = ei[N_EDGES + e];       // target (col)
    float norm = dinv[r] * ew[e] * dinv[c];
    atomicAdd(&a1[(size_t)c * HIDDEN + j], norm * h1[(size_t)r * HIDDEN + j]);
}

// ---------------------------------------------------------------------------
// 5) add self-loop term + bias, ReLU (in place into h1)
// ---------------------------------------------------------------------------
__global__ __launch_bounds__(256) void relu1_kernel(const float* __restrict__ a1,
                                                    const float* __restrict__ dinv,
                                                    const float* __restrict__ b1,
                                                    float* __restrict__ h1) {
    int t = blockIdx.x * 256 + threadIdx.x;
    if (t >= N_NODES * HIDDEN) return;
    int i = t / HIDDEN;
    int j = t - i * HIDDEN;
    float di = dinv[i];
    float v = a1[t] + di * di * h1[t] + b1[j];
    h1[t] = v > 0.0f ? v : 0.0f;
}

// ---------------------------------------------------------------------------
// 6) h2 = relu(h1) @ W2  (K=16, N=64): 4 k-steps x 4 n-tiles of f32 WMMA.
//    A fragments preloaded once per wave; W2 (4KB) in LDS. Unguarded body.
// ---------------------------------------------------------------------------
__global__ __launch_bounds__(256) void gemm2_kernel(const float* __restrict__ h1,
                                                    const float* __restrict__ W2,
                                                    float* __restrict__ h2) {
    __shared__ float sW[HIDDEN * N_CLASSES];
    int tid = threadIdx.x;
    for (int i = tid; i < HIDDEN * N_CLASSES; i += 256) sW[i] = W2[i];
    __syncthreads();

    int wave = tid >> 5;
    int lane = tid & 31;
    int tile = blockIdx.x * 8 + wave;
    if (tile >= N_TILES) return;                      // per-wave guard only
    int rowBase = tile * 16;
    int laneHalf = lane >> 4;
    int laneM = lane & 15;
    const float* hrow = h1 + (size_t)(rowBase + laneM) * HIDDEN;

    int kOff = laneHalf * 2;
    v2f a[4];
    #pragma unroll
    for (int kk = 0; kk < 4; ++kk) {
        a[kk].x = hrow[kk * 4 + kOff];
        a[kk].y = hrow[kk * 4 + kOff + 1];
    }
    float* orow = h2 + (size_t)(rowBase + laneHalf * 8) * N_CLASSES + laneM;
    #pragma unroll
    for (int nt = 0; nt < 4; ++nt) {
        int n = nt * 16 + laneM;
        v8f c = {};
        #pragma unroll
        for (int kk = 0; kk < 4; ++kk) {
            v2f b;
            b.x = sW[(kk * 4 + kOff) * N_CLASSES + n];
            b.y = sW[(kk * 4 + kOff + 1) * N_CLASSES + n];
            c = __builtin_amdgcn_wmma_f32_16x16x4_f32(false, a[kk], false, b,
                                                      (short)0, c, false, false);
        }
        #pragma unroll
        for (int r = 0; r < 8; ++r) orow[r * N_CLASSES + nt * 16] = c[r];
    }
}

// ---------------------------------------------------------------------------
// 7) scatter layer-2 messages: a2[col] += norm * h2[row]; 64 lanes per edge
// ---------------------------------------------------------------------------
__global__ __launch_bounds__(256) void scatter2_kernel(const int* __restrict__ ei,
                                                       const float* __restrict__ ew,
                                                       const float* __restrict__ dinv,
                                                       const float* __restrict__ h2,
                                                       float* __restrict__ a2) {
    int t = blockIdx.x * 256 + threadIdx.x;
    int e = t >> 6;
    int j = t & 63;
    if (e >= N_EDGES) return;
    int r = ei[e];
    int c = ei[N_EDGES + e];
    float norm = dinv[r] * ew[e] * dinv[c];
    atomicAdd(&a2[(size_t)c * N_CLASSES + j], norm * h2[(size_t)r * N_CLASSES + j]);
}

// ---------------------------------------------------------------------------
// 8) self-loop + bias + log_softmax over 64 classes; one wave32 per node
// ---------------------------------------------------------------------------
__global__ __launch_bounds__(256) void final_kernel(const float* __restrict__ a2,
                                                    const float* __restrict__ h2,
                                                    const float* __restrict__ dinv,
                                                    const float* __restrict__ b2,
                                                    float* __restrict__ out) {
    int wave = threadIdx.x >> 5;
    int lane = threadIdx.x & 31;
    int node = blockIdx.x * 8 + wave;
    if (node >= N_NODES) return;
    float di = dinv[node];
    float dd = di * di;
    size_t base = (size_t)node * N_CLASSES;
    float v0 = a2[base + lane]      + dd * h2[base + lane]      + b2[lane];
    float v1 = a2[base + 32 + lane] + dd * h2[base + 32 + lane] + b2[32 + lane];

    float m = fmaxf(v0, v1);
    #pragma unroll
    for (int off = 16; off > 0; off >>= 1) m = fmaxf(m, __shfl_xor(m, off, 32));
    float s = expf(v0 - m) + expf(v1 - m);
    #pragma unroll
    for (int off = 16; off > 0; off >>= 1) s += __shfl_xor(s, off, 32);
    float ls = logf(s);

    out[base + lane]      = v0 - m - ls;
    out[base + 32 + lane] = v1 - m - ls;
}

// ---------------------------------------------------------------------------
extern "C" void kernel_launch(void* const* d_in, const int* in_sizes, int n_in,
                              void* d_out, int out_size, void* d_ws, size_t ws_size,
                              hipStream_t stream) {
    const float* x  = (const float*)d_in[0];
    const int*   ei = (const int*)  d_in[1];   // [2, E] row-major: [0..E)=src, [E..2E)=dst
    const float* ew = (const float*)d_in[2];
    const float* W1 = (const float*)d_in[3];
    const float* b1 = (const float*)d_in[4];
    const float* W2 = (const float*)d_in[5];
    const float* b2 = (const float*)d_in[6];
    float* out = (float*)d_out;

    float* ws   = (float*)d_ws;
    float* deg  = ws;                                  // N
    float* dinv = deg  + N_NODES;                      // N
    float* h1   = dinv + N_NODES;                      // N*16
    float* a1   = h1   + (size_t)N_NODES * HIDDEN;     // N*16
    float* h2   = a1   + (size_t)N_NODES * HIDDEN;     // N*64
    float* a2   = h2   + (size_t)N_NODES * N_CLASSES;  // N*64

    const int BLK = 256;
    int initGrid = (N_NODES * N_CLASSES + BLK - 1) / BLK;
    init_kernel<<<initGrid, BLK, 0, stream>>>(deg, a1, a2);

    int degGrid = (N_EDGES + BLK - 1) / BLK;
    deg_kernel<<<degGrid, BLK, 0, stream>>>(ei + N_EDGES, ew, deg);

    int nodeGrid = (N_NODES + BLK - 1) / BLK;
    dinv_kernel<<<nodeGrid, BLK, 0, stream>>>(deg, dinv);

    int gemmGrid = (N_TILES + 7) / 8;
    gemm1_kernel<<<gemmGrid, BLK, 0, stream>>>(x, W1, h1);

    long long s1threads = (long long)N_EDGES * HIDDEN;
    int s1Grid = (int)((s1threads + BLK - 1) / BLK);
    scatter1_kernel<<<s1Grid, BLK, 0, stream>>>(ei, ew, dinv, h1, a1);

    int reluGrid = (N_NODES * HIDDEN + BLK - 1) / BLK;
    relu1_kernel<<<reluGrid, BLK, 0, stream>>>(a1, dinv, b1, h1);

    gemm2_kernel<<<gemmGrid, BLK, 0, stream>>>(h1, W2, h2);

    long long s2threads = (long long)N_EDGES * N_CLASSES;
    int s2Grid = (int)((s2threads + BLK - 1) / BLK);
    scatter2_kernel<<<s2Grid, BLK, 0, stream>>>(ei, ew, dinv, h2, a2);

    int finGrid = (N_NODES + 7) / 8;
    final_kernel<<<finGrid, BLK, 0, stream>>>(a2, h2, dinv, b2, out);
}